// DilatedAttentionNew_16320875724875
// MI455X (gfx1250) — compile-verified
//
#include <hip/hip_runtime.h>

typedef float v2f __attribute__((ext_vector_type(2)));
typedef float v8f __attribute__((ext_vector_type(8)));

#define SEG_W   2048
#define DIL_R   4
#define D_MODEL 1024
#define SEQ_N   8192
#define SEG_S   (SEQ_N / SEG_W)        // 4 segments
#define M_SUB   (SEG_W / DIL_R)        // 512 tokens per segment
#define ROWS_WG 16                     // query rows per workgroup
#define ROWBLKS (M_SUB / ROWS_WG)      // 32
#define DCHUNK  256                    // d-chunk staged in LDS for Q

// ---------------------------------------------------------------------------
// Zero the output rows that are NOT selected by the dilated index pattern.
// Selected rows are fully overwritten by the attention kernel.
// ---------------------------------------------------------------------------
__global__ __launch_bounds__(256)
void da_zero_kernel(float* __restrict__ out, const int* __restrict__ hidx) {
    const int off = hidx[0] % DIL_R;
    const int p   = blockIdx.x % SEQ_N;
    const int bb  = blockIdx.x / SEQ_N;
    const int within = p % SEG_W;
    if ((within % DIL_R) == off) return;        // attention kernel writes this row
    float4 z; z.x = 0.f; z.y = 0.f; z.z = 0.f; z.w = 0.f;
    float4* po = (float4*)(out + ((long long)bb * SEQ_N + p) * D_MODEL);
    po[threadIdx.x] = z;                        // 256 threads * float4 = 1024 floats
}

// ---------------------------------------------------------------------------
// Fused dilated self-attention for one 16-query-row tile of one (b,s) block.
//   scores = Xg Xg^T / sqrt(d)  (fp32 WMMA 16x16x4)
//   P = softmax(scores)         (LDS + shfl reductions)
//   att = P Xg                  (fp32 WMMA 16x16x4)
//   out[dilated rows] = att * (rowsum/rowsum)   // alpha == 1 (NaN-faithful)
// ---------------------------------------------------------------------------
__global__ __launch_bounds__(256)
void da_attn_kernel(const float* __restrict__ x, const int* __restrict__ hidx,
                    float* __restrict__ out) {
    __shared__ float lds_q[ROWS_WG][DCHUNK];       // 16 KB staged Q chunk
    __shared__ float lds_s[ROWS_WG][M_SUB + 8];    // ~32.5 KB scores / probs
    __shared__ float lds_red[ROWS_WG][8];          // per-wave row sums
    __shared__ float lds_alpha[ROWS_WG];

    const int off    = hidx[0] % DIL_R;
    const int wg     = blockIdx.x;
    const int rowblk = wg % ROWBLKS;
    const int blk    = wg / ROWBLKS;               // b*SEG_S + s
    const int s      = blk % SEG_S;
    const int bb     = blk / SEG_S;

    const int tid  = threadIdx.x;
    const int wv   = tid >> 5;                     // wave id 0..7
    const int lane = tid & 31;
    const int l16  = lane & 15;
    const int hi   = lane >> 4;                    // 0 or 1

    const int i0 = rowblk * ROWS_WG;               // first query row in segment
    const long long segbase = (long long)bb * SEQ_N + (long long)s * SEG_W + off;
    // token row index of sub-row i is (segbase + DIL_R*i); x/out row stride = D_MODEL

    // ---------------- Phase 1: scores = Q K^T ----------------
    v8f c[4] = {};                                 // wave's 16x64 score strip
    const int jbase = wv * 64;                     // this wave's key columns

    const float* pk[4];                            // key-row pointer per 16-col tile
#pragma unroll
    for (int jt = 0; jt < 4; ++jt) {
        const int j = jbase + jt * 16 + l16;       // key index handled by this lane
        pk[jt] = x + (segbase + (long long)DIL_R * j) * D_MODEL;
    }

    for (int dc = 0; dc < D_MODEL; dc += DCHUNK) {
        __syncthreads();                           // protect lds_q re-staging
        // stage Q[i0..i0+15][dc..dc+255] -> LDS (coalesced float4)
#pragma unroll
        for (int t = 0; t < 4; ++t) {
            const int idx = tid + 256 * t;         // 1024 float4 slots
            const int row = idx >> 6;              // 64 float4 per row
            const int c4  = idx & 63;
            const float4 qv = *(const float4*)(
                x + (segbase + (long long)DIL_R * (i0 + row)) * D_MODEL + dc + c4 * 4);
            *(float4*)&lds_q[row][c4 * 4] = qv;
        }
        __syncthreads();

        for (int k0 = 0; k0 < DCHUNK; k0 += 4) {
            // A operand: Q[l16][k0+2*hi .. +1]  (16x4 fp32 layout)
            const v2f a = *(const v2f*)&lds_q[l16][k0 + 2 * hi];
#pragma unroll
            for (int jt = 0; jt < 4; ++jt) {
                const v2f b = *(const v2f*)(pk[jt] + dc + k0 + 2 * hi);
                c[jt] = __builtin_amdgcn_wmma_f32_16x16x4_f32(
                    false, a, false, b, (short)0, c[jt], false, false);
            }
        }
    }

    // scale by 1/sqrt(D) and spill scores to LDS
    const float scale = 0.03125f;                  // 1/32
#pragma unroll
    for (int jt = 0; jt < 4; ++jt)
#pragma unroll
        for (int v = 0; v < 8; ++v)
            lds_s[v + 8 * hi][jbase + jt * 16 + l16] = c[jt][v] * scale;
    __syncthreads();

    // ---------------- softmax: 16 lanes per row ----------------
    {
        const int row = tid >> 4;                  // 0..15
        const int cl  = tid & 15;
        float mx = -3.0e38f;
#pragma unroll 4
        for (int t = 0; t < 32; ++t)
            mx = fmaxf(mx, lds_s[row][cl + 16 * t]);
#pragma unroll
        for (int m_ = 1; m_ <= 8; m_ <<= 1)
            mx = fmaxf(mx, __shfl_xor(mx, m_, 32));
        float sum = 0.f;
#pragma unroll 4
        for (int t = 0; t < 32; ++t) {
            const float e = __expf(lds_s[row][cl + 16 * t] - mx);
            lds_s[row][cl + 16 * t] = e;
            sum += e;
        }
#pragma unroll
        for (int m_ = 1; m_ <= 8; m_ <<= 1)
            sum += __shfl_xor(sum, m_, 32);
        const float inv = 1.0f / sum;
#pragma unroll 4
        for (int t = 0; t < 32; ++t)
            lds_s[row][cl + 16 * t] *= inv;
    }
    __syncthreads();

    // ---------------- Phase 2: att = P V ----------------
    v8f acc[8] = {};                               // wave's 16x128 output strip
    const int nbase = wv * 128;

    for (int k0 = 0; k0 < M_SUB; k0 += 4) {
        const v2f a = *(const v2f*)&lds_s[l16][k0 + 2 * hi];   // P[l16][k..]
        const float* vr0 = x + (segbase + (long long)DIL_R * (k0 + 2 * hi)) * D_MODEL;
        const float* vr1 = vr0 + (long long)DIL_R * D_MODEL;
#pragma unroll
        for (int nt = 0; nt < 8; ++nt) {
            const int col = nbase + nt * 16 + l16;
            v2f b;
            b.x = vr0[col];
            b.y = vr1[col];
            acc[nt] = __builtin_amdgcn_wmma_f32_16x16x4_f32(
                false, a, false, b, (short)0, acc[nt], false, false);
        }
    }

    // ---------------- row sums -> alpha = d/d (== 1, NaN-faithful) ----------------
    float s8[8];
#pragma unroll
    for (int v = 0; v < 8; ++v) {
        float t = 0.f;
#pragma unroll
        for (int nt = 0; nt < 8; ++nt) t += acc[nt][v];
#pragma unroll
        for (int m_ = 1; m_ <= 8; m_ <<= 1) t += __shfl_xor(t, m_, 32);
        s8[v] = t;
    }
    if (l16 == 0) {
#pragma unroll
        for (int v = 0; v < 8; ++v) lds_red[v + 8 * hi][wv] = s8[v];
    }
    __syncthreads();
    if (tid < 16) {
        float dnm = 0.f;
#pragma unroll
        for (int w_ = 0; w_ < 8; ++w_) dnm += lds_red[tid][w_];
        lds_alpha[tid] = dnm / dnm;                // exactly 1.0 unless rowsum==0
    }
    __syncthreads();

    // ---------------- scatter-write output rows ----------------
#pragma unroll
    for (int v = 0; v < 8; ++v) {
        const int irow = i0 + v + 8 * hi;          // row in segment
        float* po = out + (segbase + (long long)DIL_R * irow) * D_MODEL;
        const float al = lds_alpha[v + 8 * hi];
#pragma unroll
        for (int nt = 0; nt < 8; ++nt)
            po[nbase + nt * 16 + l16] = acc[nt][v] * al;
    }
}

// ---------------------------------------------------------------------------
extern "C" void kernel_launch(void* const* d_in, const int* in_sizes, int n_in,
                              void* d_out, int out_size, void* d_ws, size_t ws_size,
                              hipStream_t stream) {
    const float* x    = (const float*)d_in[0];
    const int*   hidx = (const int*)d_in[1];
    float*       out  = (float*)d_out;

    const int nb = in_sizes[0] / (SEQ_N * D_MODEL);   // batch (4)

    da_zero_kernel<<<dim3(nb * SEQ_N), dim3(256), 0, stream>>>(out, hidx);
    da_attn_kernel<<<dim3(nb * SEG_S * ROWBLKS), dim3(256), 0, stream>>>(x, hidx, out);
}